// CCM_5609227289137
// MI455X (gfx1250) — compile-verified
//
#include <hip/hip_runtime.h>
#include <hip/hip_bf16.h>
#include <stdint.h>

// Problem dims (fixed by the reference)
#define BB 8
#define TT_DIM 1000
#define FF_DIM 257
#define TCHUNK 16      // 16 x 8B out elements = one 128B line per thread per chunk
#define NROWS (TCHUNK + 2)
#define CP 260         // padded row pitch (float2 elements), F+2=259 rounded up
#define RP (TT_DIM + 2)

// ---------------------------------------------------------------------------
// Kernel A: build zero-padded, transposed complex spectrogram in workspace.
// xpad[b][row][col] = (row>=2 && 1<=col<=F) ? x[b, col-1, row-2, :] : 0
// Reads are coalesced along t (threadIdx.x -> row).
// ---------------------------------------------------------------------------
__global__ __launch_bounds__(1024)
void build_xpad_kernel(const float* __restrict__ x, float2* __restrict__ xpad) {
    const int col = blockIdx.x % (FF_DIM + 2);
    const int b   = blockIdx.x / (FF_DIM + 2);
    const int row = threadIdx.x;
    if (row >= RP) return;
    float2 v; v.x = 0.0f; v.y = 0.0f;
    const int t = row - 2;
    const int f = col - 1;
    if (t >= 0 && f >= 0 && f < FF_DIM) {
        const float* p = x + (((size_t)b * FF_DIM + f) * TT_DIM + t) * 2;
        v.x = p[0];
        v.y = p[1];
    }
    xpad[((size_t)b * RP + row) * CP + col] = v;
}

// ---------------------------------------------------------------------------
// Kernel B: fused mask-build + 3x3 complex stencil.
// One thread per frequency bin (288-thread blocks = 9 wave32), looping over a
// 16-deep time chunk. The full 18-row padded halo window for the chunk is
// staged into LDS with one burst of global_load_async_to_lds_b64, followed by
// a single s_wait_asynccnt + barrier; the 16 compute steps then run sync-free.
// ---------------------------------------------------------------------------
template<bool USE_WS>
__global__ __launch_bounds__(288)
void ccm_stencil_kernel(const float* __restrict__ m,
                        const float* __restrict__ x,
                        const float2* __restrict__ xpad,
                        float* __restrict__ out,
                        int nChunk) {
    __shared__ float2 rows[NROWS][CP];

    const int tid   = threadIdx.x;
    const int b     = blockIdx.x / nChunk;
    const int chunk = blockIdx.x % nChunk;
    const int t0    = chunk * TCHUNK;
    const int tEnd  = (t0 + TCHUNK < TT_DIM) ? (t0 + TCHUNK) : TT_DIM;
    const int nRows = (tEnd - t0) + 2;          // <= 18; rows t0 .. tEnd+1 <= T+1
    const size_t planeTF = (size_t)TT_DIM * FF_DIM;

    if (USE_WS) {
        // One async DMA burst: lane `tid` streams element `tid` of each padded
        // row straight into LDS (ASYNCcnt-tracked, no VGPR round trip).
        if (tid < FF_DIM + 2) {
            const float2* g0 = xpad + ((size_t)b * RP + t0) * CP + tid;
            for (int r = 0; r < nRows; ++r) {
                unsigned lds_off = (unsigned)(uintptr_t)(&rows[r][tid]);
                const float2* g = g0 + (size_t)r * CP;
                asm volatile("global_load_async_to_lds_b64 %0, %1, off"
                             :: "v"(lds_off), "v"(g) : "memory");
            }
        }
        asm volatile("s_wait_asynccnt 0x0" ::: "memory");
    } else {
        // Fallback (ws too small): direct strided loads with inline pad/transpose.
        if (tid < FF_DIM + 2) {
            const int f = tid - 1;
            for (int r = 0; r < nRows; ++r) {
                const int t = t0 + r - 2;
                float2 v; v.x = 0.0f; v.y = 0.0f;
                if (t >= 0 && f >= 0 && f < FF_DIM) {
                    const float* p = x + (((size_t)b * FF_DIM + f) * TT_DIM + t) * 2;
                    v.x = p[0];
                    v.y = p[1];
                }
                rows[r][tid] = v;
            }
        }
    }
    __syncthreads();   // whole halo window staged; compute runs sync-free below

    if (tid < FF_DIM) {
        const float SQ32 = 0.86602540378443864676f;
        const float* mb0 = m + (size_t)b * 27 * planeTF + (size_t)t0 * FF_DIM + tid;
        for (int t = t0; t < tEnd; ++t) {
            const float* mb = mb0 + (size_t)(t - t0) * FF_DIM;
            float accRe = 0.0f, accIm = 0.0f;
#pragma unroll
            for (int mm = 0; mm < 3; ++mm) {
                const float2* rowp = rows[t - t0 + mm];
#pragma unroll
                for (int nn = 0; nn < 3; ++nn) {
                    const int c = mm * 3 + nn;
                    // streaming, non-temporal: keep the 222MB m tensor out of L2
                    float m0 = __builtin_nontemporal_load(mb + (size_t)c * planeTF);
                    float m1 = __builtin_nontemporal_load(mb + (size_t)(c + 9) * planeTF);
                    float m2 = __builtin_nontemporal_load(mb + (size_t)(c + 18) * planeTF);
                    // mask entry: Re = m0 - 0.5(m1+m2), Im = (sqrt(3)/2)(m1-m2)
                    float Mre = fmaf(-0.5f, m1 + m2, m0);
                    float Mim = SQ32 * (m1 - m2);
                    float2 xv = rowp[tid + nn];           // ds_load_b64, conflict-free
                    accRe = fmaf(Mre, xv.x, accRe);
                    accRe = fmaf(-Mim, xv.y, accRe);
                    accIm = fmaf(Mre, xv.y, accIm);
                    accIm = fmaf(Mim, xv.x, accIm);
                }
            }
            float* o = out + (((size_t)b * FF_DIM + tid) * TT_DIM + t) * 2;
            o[0] = accRe;   // adjacent pair -> global_store_b64; 16 consecutive t
            o[1] = accIm;   // per thread fill one 128B line in L2
        }
    }
}

// ---------------------------------------------------------------------------
extern "C" void kernel_launch(void* const* d_in, const int* in_sizes, int n_in,
                              void* d_out, int out_size, void* d_ws, size_t ws_size,
                              hipStream_t stream) {
    (void)in_sizes; (void)n_in; (void)out_size;
    const float* m = (const float*)d_in[0];   // (B, 27, T, F) fp32
    const float* x = (const float*)d_in[1];   // (B, F, T, 2) fp32
    float* out = (float*)d_out;               // (B, F, T, 2) fp32

    const int nChunk = (TT_DIM + TCHUNK - 1) / TCHUNK;          // 63
    const size_t need = (size_t)BB * RP * CP * sizeof(float2);  // ~16.7 MB

    dim3 gridS(BB * nChunk);   // 504 blocks
    dim3 blockS(288);          // 9 wave32 (257 active freq lanes + halo staging)

    if (ws_size >= need) {
        float2* xpad = (float2*)d_ws;
        build_xpad_kernel<<<dim3(BB * (FF_DIM + 2)), dim3(1024), 0, stream>>>(x, xpad);
        ccm_stencil_kernel<true><<<gridS, blockS, 0, stream>>>(m, x, xpad, out, nChunk);
    } else {
        ccm_stencil_kernel<false><<<gridS, blockS, 0, stream>>>(m, x, nullptr, out, nChunk);
    }
}